// ASAP_Pooling_55860344652297
// MI455X (gfx1250) — compile-verified
//
#include <hip/hip_runtime.h>

#define NEG_SLOPE 0.2f

typedef float v2f __attribute__((ext_vector_type(2)));
typedef float v8f __attribute__((ext_vector_type(8)));

// ---------------- helpers ----------------

__device__ __forceinline__ void edge_get(const int* __restrict__ src, const int* __restrict__ dst,
                                         const float* __restrict__ ew, int E, int e,
                                         int& r, int& c, float& w) {
  if (e < E) { r = src[e]; c = dst[e]; w = ew[e]; }
  else       { r = e - E;  c = r;      w = 1.0f;  }
}

__device__ __forceinline__ int edge_row(const int* __restrict__ src, int E, int e) {
  return (e < E) ? src[e] : (e - E);
}

__device__ __forceinline__ float wave_sum(float v) {
#pragma unroll
  for (int off = 16; off > 0; off >>= 1) v += __shfl_xor(v, off, 32);
  return v;
}

// exact float atomic-max via int/uint ordering trick (init value must be -inf)
__device__ __forceinline__ void atomicMaxF(float* addr, float val) {
  if (val >= 0.0f) atomicMax((int*)addr, __float_as_int(val));
  else             atomicMin((unsigned int*)addr, __float_as_uint(val));
}

// ---------------- fills ----------------

__global__ void fill_f32(float* __restrict__ p, float v, int n) {
  for (int i = blockIdx.x * blockDim.x + threadIdx.x; i < n; i += gridDim.x * blockDim.x) p[i] = v;
}
__global__ void fill_i32(int* __restrict__ p, int v, int n) {
  for (int i = blockIdx.x * blockDim.x + threadIdx.x; i < n; i += gridDim.x * blockDim.x) p[i] = v;
}

// ---------------- graph prep ----------------

__global__ void edge_prep(const int* __restrict__ src, const int* __restrict__ dst,
                          const float* __restrict__ ew, int E, int N,
                          float* __restrict__ deg, int* __restrict__ cnt) {
  int e = blockIdx.x * blockDim.x + threadIdx.x;
  if (e >= E + N) return;
  int r, c; float w; edge_get(src, dst, ew, E, e, r, c, w);
  atomicAdd(&deg[r], w);
  atomicAdd(&cnt[c], 1);
}

__global__ void dinv_k(const float* __restrict__ deg, float* __restrict__ dinv, int N) {
  int i = blockIdx.x * blockDim.x + threadIdx.x;
  if (i < N) dinv[i] = (deg[i] > 0.0f) ? rsqrtf(deg[i]) : 0.0f;
}

// ---------------- WMMA f32 GEMM: D[M,C] = A[M,C] * W[C,C] (+bias) ----------------
// one wave per 16x16 output tile; K-loop in steps of 4 using V_WMMA_F32_16X16X4_F32.
// A frag (16x4): lanes 0-15 hold {K=k,k+1}, lanes 16-31 hold {K=k+2,k+3} (ISA 7.12.2)
// B frag (4x16): assumed mirrored row striping.
// C/D: VGPR g -> row (g + 8*half), col = lane&15.

__global__ __launch_bounds__(32) void gemm_wmma_f32(const float* __restrict__ A,
                                                    const float* __restrict__ W,
                                                    const float* __restrict__ bias,
                                                    float* __restrict__ D, int C) {
  const int tm = blockIdx.x, tn = blockIdx.y;
  const int lane = threadIdx.x;
  const int half = lane >> 4;
  const int l15  = lane & 15;
  v8f acc = {};
  const float* arow = A + (size_t)(tm * 16 + l15) * C;
  for (int k = 0; k < C; k += 4) {
    v2f a, b;
    a.x = arow[k + half * 2 + 0];
    a.y = arow[k + half * 2 + 1];
    const float* wp = W + (size_t)(k + half * 2) * C + tn * 16 + l15;
    b.x = wp[0];
    b.y = wp[C];
    acc = __builtin_amdgcn_wmma_f32_16x16x4_f32(false, a, false, b, (short)0, acc, false, false);
  }
  const float bv = bias ? bias[tn * 16 + l15] : 0.0f;
#pragma unroll
  for (int g = 0; g < 8; ++g) {
    int row = tm * 16 + half * 8 + g;
    D[(size_t)row * C + tn * 16 + l15] = acc[g] + bv;
  }
}

// ---------------- GCN scatter: xp[row,:] += norm * h[col,:] ----------------

__global__ void scatter_xpool(const int* __restrict__ src, const int* __restrict__ dst,
                              const float* __restrict__ ew, int E, int N,
                              const float* __restrict__ dinv, const float* __restrict__ h,
                              float* __restrict__ xp, int C) {
  int gw = (blockIdx.x * blockDim.x + threadIdx.x) >> 5;
  int lane = threadIdx.x & 31;
  if (gw >= E + N) return;
  int r, c; float w; edge_get(src, dst, ew, E, gw, r, c, w);
  float norm = dinv[r] * w * dinv[c];
  const float* hc = h + (size_t)c * C;
  float* xr = xp + (size_t)r * C;
  for (int ch = lane; ch < C; ch += 32) atomicAdd(&xr[ch], norm * hc[ch]);
}

__global__ void add_bias(float* __restrict__ xp, const float* __restrict__ b, int N, int C) {
  int i = blockIdx.x * blockDim.x + threadIdx.x;
  if (i < N * C) xp[i] += b[i % C];
}

// ---------------- segment max of x_pool[col] over row ----------------

__global__ void scatter_xq(const int* __restrict__ src, const int* __restrict__ dst,
                           const float* __restrict__ ew, int E, int N,
                           const float* __restrict__ xp, float* __restrict__ xq, int C) {
  int gw = (blockIdx.x * blockDim.x + threadIdx.x) >> 5;
  int lane = threadIdx.x & 31;
  if (gw >= E + N) return;
  int r, c; float w; edge_get(src, dst, ew, E, gw, r, c, w);
  const float* xc = xp + (size_t)c * C;
  float* qr = xq + (size_t)r * C;
  for (int ch = lane; ch < C; ch += 32) atomicMaxF(&qr[ch], xc[ch]);
}

// ---------------- per-node dots: q1 = Mq . Wa[:C], p2 = xp . Wa[C:] ----------------

__global__ void node_qp(const float* __restrict__ mq, const float* __restrict__ xp,
                        const float* __restrict__ Wa, float* __restrict__ q1,
                        float* __restrict__ p2, int N, int C) {
  int gw = (blockIdx.x * blockDim.x + threadIdx.x) >> 5;
  int lane = threadIdx.x & 31;
  if (gw >= N) return;
  float s1 = 0.0f, s2 = 0.0f;
  const float* mr = mq + (size_t)gw * C;
  const float* xr = xp + (size_t)gw * C;
  for (int ch = lane; ch < C; ch += 32) {
    s1 += mr[ch] * Wa[ch];
    s2 += xr[ch] * Wa[C + ch];
  }
  s1 = wave_sum(s1); s2 = wave_sum(s2);
  if (lane == 0) { q1[gw] = s1; p2[gw] = s2; }
}

// ---------------- edge scores + segment softmax ----------------

__global__ void edge_score(const int* __restrict__ src, const int* __restrict__ dst,
                           const float* __restrict__ ew, int E, int N,
                           const float* __restrict__ q1, const float* __restrict__ p2,
                           const float* __restrict__ ba, float* __restrict__ s,
                           float* __restrict__ mrow) {
  int e = blockIdx.x * blockDim.x + threadIdx.x;
  if (e >= E + N) return;
  int r, c; float w; edge_get(src, dst, ew, E, e, r, c, w);
  float v = q1[r] + p2[c] + ba[0];
  v = (v >= 0.0f) ? v : NEG_SLOPE * v;
  s[e] = v;
  atomicMaxF(&mrow[r], v);
}

__global__ void edge_exp(const int* __restrict__ src, const int* __restrict__ dst,
                         const float* __restrict__ ew, int E, int N,
                         const float* __restrict__ s, const float* __restrict__ mrow,
                         float* __restrict__ eexp, float* __restrict__ den) {
  int e = blockIdx.x * blockDim.x + threadIdx.x;
  if (e >= E + N) return;
  int r = edge_row(src, E, e);
  float v = __expf(s[e] - mrow[r]);
  eexp[e] = v;
  atomicAdd(&den[r], v);
}

__global__ void edge_att(const int* __restrict__ src, const int* __restrict__ dst,
                         const float* __restrict__ ew, int E, int N,
                         const float* __restrict__ eexp, const float* __restrict__ den,
                         float* __restrict__ att) {
  int e = blockIdx.x * blockDim.x + threadIdx.x;
  if (e >= E + N) return;
  int r = edge_row(src, E, e);
  att[e] = eexp[e] / (den[r] + 1e-16f);
}

// ---------------- out[row,:] += att * x[col,:] ----------------

__global__ void scatter_out(const int* __restrict__ src, const int* __restrict__ dst,
                            const float* __restrict__ ew, int E, int N,
                            const float* __restrict__ att, const float* __restrict__ x,
                            float* __restrict__ outc, int C) {
  int gw = (blockIdx.x * blockDim.x + threadIdx.x) >> 5;
  int lane = threadIdx.x & 31;
  if (gw >= E + N) return;
  int r, c; float w; edge_get(src, dst, ew, E, gw, r, c, w);
  float a = att[gw];
  const float* xc = x + (size_t)c * C;
  float* orr = outc + (size_t)r * C;
  for (int ch = lane; ch < C; ch += 32) atomicAdd(&orr[ch], a * xc[ch]);
}

// ---------------- LEConv per-node dots ----------------

__global__ void le_node(const float* __restrict__ outc, const float* __restrict__ Wle,
                        const float* __restrict__ Wl1, const float* __restrict__ bl1,
                        const float* __restrict__ Wl2, const float* __restrict__ bl2,
                        float* __restrict__ hL, float* __restrict__ f1v,
                        float* __restrict__ f2v, int N, int C) {
  int gw = (blockIdx.x * blockDim.x + threadIdx.x) >> 5;
  int lane = threadIdx.x & 31;
  if (gw >= N) return;
  const float* orr = outc + (size_t)gw * C;
  float s0 = 0.0f, s1 = 0.0f, s2 = 0.0f;
  for (int ch = lane; ch < C; ch += 32) {
    float v = orr[ch];
    s0 += v * Wle[ch];
    s1 += v * Wl1[ch];
    s2 += v * Wl2[ch];
  }
  s0 = wave_sum(s0); s1 = wave_sum(s1); s2 = wave_sum(s2);
  if (lane == 0) { hL[gw] = s0; f1v[gw] = s1 + bl1[0]; f2v[gw] = s2 + bl2[0]; }
}

__global__ void le_edge(const int* __restrict__ src, const int* __restrict__ dst,
                        const float* __restrict__ ew, int E, int N,
                        const float* __restrict__ hL, float* __restrict__ dg2,
                        float* __restrict__ agg) {
  int e = blockIdx.x * blockDim.x + threadIdx.x;
  if (e >= E + N) return;
  int r, c; float w; edge_get(src, dst, ew, E, e, r, c, w);
  if (r == c) return;  // LEConv removes self loops
  atomicAdd(&dg2[r], w);
  atomicAdd(&agg[r], w * hL[c]);
}

__global__ void fitness_k(const float* __restrict__ dg2, const float* __restrict__ f1v,
                          const float* __restrict__ agg, const float* __restrict__ f2v,
                          float* __restrict__ fit, int N) {
  int i = blockIdx.x * blockDim.x + threadIdx.x;
  if (i >= N) return;
  float z = dg2[i] * f1v[i] + agg[i] + f2v[i];
  fit[i] = 1.0f / (1.0f + __expf(-z));
}

// ---------------- top-k via single-workgroup LDS bitonic sort ----------------
// order: descending fitness, ascending index on ties (jax.lax.top_k semantics)

#define SORT_N 8192

__global__ __launch_bounds__(1024) void topk_k(const float* __restrict__ fit, int N, int kN,
                                               int* __restrict__ permi, int* __restrict__ inperm,
                                               int* __restrict__ nidx, float* __restrict__ permf) {
  __shared__ float key[SORT_N];
  __shared__ int   idx[SORT_N];
  const int t = threadIdx.x;
  for (int i = t; i < SORT_N; i += 1024) {
    key[i] = (i < N) ? fit[i] : -__builtin_huge_valf();
    idx[i] = i;
  }
  __syncthreads();
  for (int k = 2; k <= SORT_N; k <<= 1) {
    for (int j = k >> 1; j > 0; j >>= 1) {
      for (int i = t; i < SORT_N; i += 1024) {
        int ixj = i ^ j;
        if (ixj > i) {
          bool dir = ((i & k) == 0);
          float ka = key[i], kb = key[ixj];
          int ia = idx[i], ib = idx[ixj];
          bool before = (ka > kb) || (ka == kb && ia < ib);
          if (before != dir) { key[i] = kb; key[ixj] = ka; idx[i] = ib; idx[ixj] = ia; }
        }
      }
      __syncthreads();
    }
  }
  for (int i = t; i < N; i += 1024) inperm[i] = 0;
  __syncthreads();
  for (int i = t; i < kN; i += 1024) {
    int p = idx[i];
    permi[i] = p;
    permf[i] = (float)p;
    inperm[p] = 1;
    nidx[p] = i;
  }
}

__global__ void xout_k(const float* __restrict__ outc, const float* __restrict__ fit,
                       const int* __restrict__ permi, float* __restrict__ xout, int kN, int C) {
  int t = blockIdx.x * blockDim.x + threadIdx.x;
  if (t >= kN * C) return;
  int i = t / C, ch = t % C;
  int p = permi[i];
  xout[t] = outc[(size_t)p * C + ch] * fit[p];
}

__global__ void sval_k(const int* __restrict__ src, const int* __restrict__ dst,
                       const float* __restrict__ ew, int E, int N,
                       const int* __restrict__ inperm, const float* __restrict__ att,
                       float* __restrict__ sval) {
  int e = blockIdx.x * blockDim.x + threadIdx.x;
  if (e >= E + N) return;
  int r = edge_row(src, E, e);
  sval[e] = inperm[r] ? att[e] : 0.0f;
}

// ---------------- CSR by column + sparse S^T A S ----------------

__global__ __launch_bounds__(1024) void scan_k(const int* __restrict__ cnt, int* __restrict__ colptr,
                                               int* __restrict__ cursor, int N) {
  __shared__ int part[1024];
  const int t = threadIdx.x;
  const int chunk = N / 1024;  // N must be a multiple of 1024 (8192 -> 8)
  int local[16];
  int s = 0;
  int base = t * chunk;
  for (int i = 0; i < chunk; ++i) { local[i] = s; s += cnt[base + i]; }
  part[t] = s;
  __syncthreads();
  for (int off = 1; off < 1024; off <<= 1) {
    int v = (t >= off) ? part[t - off] : 0;
    __syncthreads();
    part[t] += v;
    __syncthreads();
  }
  int offset = (t == 0) ? 0 : part[t - 1];
  for (int i = 0; i < chunk; ++i) {
    int v = offset + local[i];
    colptr[base + i] = v;
    cursor[base + i] = v;
  }
  if (t == 1023) colptr[N] = part[1023];
}

__global__ void csr_scatter(const int* __restrict__ src, const int* __restrict__ dst,
                            const float* __restrict__ ew, int E, int N,
                            int* __restrict__ cursor, int* __restrict__ collist) {
  int e = blockIdx.x * blockDim.x + threadIdx.x;
  if (e >= E + N) return;
  int r, c; float w; edge_get(src, dst, ew, E, e, r, c, w);
  int pos = atomicAdd(&cursor[c], 1);
  collist[pos] = e;
}

// Emat[a,b] += Sd[row_e2,a] * w_e2 * Sd[col_e2,b]; Sd nonzeros per node m are
// edges e1 with col(e1)==m and S_val(e1)!=0 -> a=nidx[row(e1)], val=S_val(e1).
__global__ void emat_k(const int* __restrict__ src, const int* __restrict__ dst,
                       const float* __restrict__ ew, int E, int N,
                       const int* __restrict__ colptr, const int* __restrict__ collist,
                       const int* __restrict__ nidx, const float* __restrict__ sval,
                       float* __restrict__ emat, int kN) {
  int e2 = blockIdx.x * blockDim.x + threadIdx.x;
  if (e2 >= E + N) return;
  int r2, c2; float w2; edge_get(src, dst, ew, E, e2, r2, c2, w2);
  int jb = colptr[r2], je = colptr[r2 + 1];
  int kb = colptr[c2], ke = colptr[c2 + 1];
  for (int p = jb; p < je; ++p) {
    int e1 = collist[p];
    float s1 = sval[e1];
    if (s1 == 0.0f) continue;
    int a = nidx[edge_row(src, E, e1)];
    float s1w = s1 * w2;
    for (int q = kb; q < ke; ++q) {
      int e3 = collist[q];
      float s3 = sval[e3];
      if (s3 == 0.0f) continue;
      int b = nidx[edge_row(src, E, e3)];
      atomicAdd(&emat[(size_t)a * kN + b], s1w * s3);
    }
  }
}

__global__ void diag_k(float* __restrict__ emat, int kN) {
  int i = blockIdx.x * blockDim.x + threadIdx.x;
  if (i < kN) emat[(size_t)i * kN + i] = 1.0f;
}

// ---------------- launcher ----------------

static inline int cdiv(int a, int b) { return (a + b - 1) / b; }

extern "C" void kernel_launch(void* const* d_in, const int* in_sizes, int n_in,
                              void* d_out, int out_size, void* d_ws, size_t ws_size,
                              hipStream_t stream) {
  const float* x   = (const float*)d_in[0];
  const int*   esr = (const int*)d_in[1];
  const int*   eds = (const int*)d_in[2];
  const float* ew  = (const float*)d_in[3];
  const float* Wg  = (const float*)d_in[4];
  const float* bg  = (const float*)d_in[5];
  const float* Wq  = (const float*)d_in[6];
  const float* bq  = (const float*)d_in[7];
  const float* Wa  = (const float*)d_in[8];
  const float* ba  = (const float*)d_in[9];
  const float* Wle = (const float*)d_in[10];
  const float* Wl1 = (const float*)d_in[11];
  const float* bl1 = (const float*)d_in[12];
  const float* Wl2 = (const float*)d_in[13];
  const float* bl2 = (const float*)d_in[14];

  const int C    = in_sizes[5];            // 128
  const int N    = in_sizes[0] / C;        // 8192
  const int E    = in_sizes[1];            // 131072
  const int Etot = E + N;                  // 139264
  const int kN   = (N + 3) / 4;            // 2048

  // workspace layout
  float* ws = (float*)d_ws;
  size_t o = 0;
  float* h    = ws + o; o += (size_t)N * C;
  float* xp   = ws + o; o += (size_t)N * C;
  float* xq   = ws + o; o += (size_t)N * C;
  float* mq   = ws + o; o += (size_t)N * C;
  float* outc = ws + o; o += (size_t)N * C;
  float* deg  = ws + o; o += N;
  float* dinv = ws + o; o += N;
  float* q1   = ws + o; o += N;
  float* p2   = ws + o; o += N;
  float* mrow = ws + o; o += N;
  float* den  = ws + o; o += N;
  float* dg2  = ws + o; o += N;
  float* agg  = ws + o; o += N;
  float* hL   = ws + o; o += N;
  float* f1v  = ws + o; o += N;
  float* f2v  = ws + o; o += N;
  float* fit  = ws + o; o += N;
  float* sarr = ws + o; o += Etot;
  float* eexp = ws + o; o += Etot;
  int* ib = (int*)(ws + o);
  size_t oi = 0;
  int* cnt     = ib + oi; oi += N;
  int* colptr  = ib + oi; oi += N + 1;
  int* cursor  = ib + oi; oi += N;
  int* collist = ib + oi; oi += Etot;
  int* nidx    = ib + oi; oi += N;
  int* inperm  = ib + oi; oi += N;
  int* permi   = ib + oi; oi += kN;

  // output layout: x_out [kN*C] | Emat [kN*kN] | perm [kN] | S_val [Etot] | att [Etot]
  float* out   = (float*)d_out;
  float* xout  = out;
  float* emat  = xout + (size_t)kN * C;
  float* permf = emat + (size_t)kN * kN;
  float* svalo = permf + kN;
  float* atto  = svalo + Etot;

  const int TB = 256;
  const float NEG_INF = -__builtin_huge_valf();

  // init
  fill_f32<<<512, TB, 0, stream>>>(xp, 0.0f, N * C);
  fill_f32<<<512, TB, 0, stream>>>(outc, 0.0f, N * C);
  fill_f32<<<512, TB, 0, stream>>>(xq, NEG_INF, N * C);
  fill_f32<<<cdiv(N, TB), TB, 0, stream>>>(deg, 0.0f, N);
  fill_f32<<<cdiv(N, TB), TB, 0, stream>>>(den, 0.0f, N);
  fill_f32<<<cdiv(N, TB), TB, 0, stream>>>(dg2, 0.0f, N);
  fill_f32<<<cdiv(N, TB), TB, 0, stream>>>(agg, 0.0f, N);
  fill_f32<<<cdiv(N, TB), TB, 0, stream>>>(mrow, NEG_INF, N);
  fill_i32<<<cdiv(N, TB), TB, 0, stream>>>(cnt, 0, N);

  // GCNConv
  edge_prep<<<cdiv(Etot, TB), TB, 0, stream>>>(esr, eds, ew, E, N, deg, cnt);
  dinv_k<<<cdiv(N, TB), TB, 0, stream>>>(deg, dinv, N);
  gemm_wmma_f32<<<dim3(N / 16, C / 16), 32, 0, stream>>>(x, Wg, nullptr, h, C);
  scatter_xpool<<<cdiv(Etot * 32, TB), TB, 0, stream>>>(esr, eds, ew, E, N, dinv, h, xp, C);
  add_bias<<<cdiv(N * C, TB), TB, 0, stream>>>(xp, bg, N, C);

  // master query attention
  scatter_xq<<<cdiv(Etot * 32, TB), TB, 0, stream>>>(esr, eds, ew, E, N, xp, xq, C);
  gemm_wmma_f32<<<dim3(N / 16, C / 16), 32, 0, stream>>>(xq, Wq, bq, mq, C);
  node_qp<<<cdiv(N * 32, TB), TB, 0, stream>>>(mq, xp, Wa, q1, p2, N, C);
  edge_score<<<cdiv(Etot, TB), TB, 0, stream>>>(esr, eds, ew, E, N, q1, p2, ba, sarr, mrow);
  edge_exp<<<cdiv(Etot, TB), TB, 0, stream>>>(esr, eds, ew, E, N, sarr, mrow, eexp, den);
  edge_att<<<cdiv(Etot, TB), TB, 0, stream>>>(esr, eds, ew, E, N, eexp, den, atto);
  scatter_out<<<cdiv(Etot * 32, TB), TB, 0, stream>>>(esr, eds, ew, E, N, atto, x, outc, C);

  // LEConv fitness
  le_node<<<cdiv(N * 32, TB), TB, 0, stream>>>(outc, Wle, Wl1, bl1, Wl2, bl2, hL, f1v, f2v, N, C);
  le_edge<<<cdiv(Etot, TB), TB, 0, stream>>>(esr, eds, ew, E, N, hL, dg2, agg);
  fitness_k<<<cdiv(N, TB), TB, 0, stream>>>(dg2, f1v, agg, f2v, fit, N);

  // top-k + gathered outputs
  topk_k<<<1, 1024, 0, stream>>>(fit, N, kN, permi, inperm, nidx, permf);
  xout_k<<<cdiv(kN * C, TB), TB, 0, stream>>>(outc, fit, permi, xout, kN, C);
  sval_k<<<cdiv(Etot, TB), TB, 0, stream>>>(esr, eds, ew, E, N, inperm, atto, svalo);

  // graph connectivity: sparse S^T A S
  scan_k<<<1, 1024, 0, stream>>>(cnt, colptr, cursor, N);
  csr_scatter<<<cdiv(Etot, TB), TB, 0, stream>>>(esr, eds, ew, E, N, cursor, collist);
  fill_f32<<<1024, TB, 0, stream>>>(emat, 0.0f, kN * kN);
  emat_k<<<cdiv(Etot, TB), TB, 0, stream>>>(esr, eds, ew, E, N, colptr, collist, nidx, svalo, emat, kN);
  diag_k<<<cdiv(kN, TB), TB, 0, stream>>>(emat, kN);
}